// Proposal_128849018983
// MI455X (gfx1250) — compile-verified
//
#include <hip/hip_runtime.h>
#include <stdint.h>
#include <math.h>

#define A_COUNT 90000
#define NB      32
#define PRE_N   6000
#define POST_N  300
#define NMS_THR 0.7f
#define IMG_MAX 1599.0f
#define SORT_N  8192          // next pow2 >= PRE_N (+ room for boundary ties)
#define NW      188           // ceil(PRE_N / 32)
#define BLK     1024
#define DEPTH   4             // async-to-LDS pipeline depth per wave

// ---------------------------------------------------------------------------
// CDNA5 async-to-LDS path (gfx1250): per-lane DMA of one dword from global
// into this lane's LDS slot, tracked by ASYNCcnt; loads complete in order.
// ---------------------------------------------------------------------------
__device__ __forceinline__ void async_f32_to_lds(uint32_t lds_byte_off,
                                                 const float* gaddr) {
#if defined(__gfx1250__)
  asm volatile("global_load_async_to_lds_b32 %0, %1, off"
               :: "v"(lds_byte_off), "v"(gaddr)
               : "memory");
#else
  (void)lds_byte_off; (void)gaddr;
#endif
}

__device__ __forceinline__ void wait_async_le(int n) {
#if defined(__gfx1250__)
  if (n <= 0)      asm volatile("s_wait_asynccnt 0" ::: "memory");
  else if (n == 1) asm volatile("s_wait_asynccnt 1" ::: "memory");
  else if (n == 2) asm volatile("s_wait_asynccnt 2" ::: "memory");
  else             asm volatile("s_wait_asynccnt 3" ::: "memory");
#else
  (void)n;
#endif
}

// Stream this batch's 90000 scores (stride-2 floats) through a DEPTH-deep
// async-to-LDS ring; calls consume(g, scorebits) for every element.
// No cross-thread sharing of stage slots => no barriers needed inside.
template <typename F>
__device__ __forceinline__ void stream_scores(const float* __restrict__ sb,
                                              float (*stage)[BLK],
                                              int tid, F&& consume) {
  const int NCH = (A_COUNT + BLK - 1) / BLK;   // 88 chunks
  for (int c = 0; c < DEPTH && c < NCH; ++c) {
    int g = c * BLK + tid;
    if (g >= A_COUNT) g = A_COUNT - 1;         // clamp: uniform issue per wave
    async_f32_to_lds((uint32_t)(uintptr_t)&stage[c][tid], sb + (size_t)g * 2);
  }
  for (int c = 0; c < NCH; ++c) {
    const int out = (NCH - c < DEPTH) ? (NCH - c) : DEPTH;  // outstanding
    wait_async_le(out - 1);                    // retires chunk c (in-order)
    const int g = c * BLK + tid;
    if (g < A_COUNT) consume(g, __float_as_uint(stage[c % DEPTH][tid]));
    const int cn = c + DEPTH;
    if (cn < NCH) {
      int gn = cn * BLK + tid;
      if (gn >= A_COUNT) gn = A_COUNT - 1;
      async_f32_to_lds((uint32_t)(uintptr_t)&stage[c % DEPTH][tid],
                       sb + (size_t)gn * 2);
    }
  }
}

// ---------------------------------------------------------------------------
// Fused per-batch kernel: radix-select -> bitonic sort -> decode -> greedy NMS,
// everything resident in CDNA5's 320KB LDS (uses ~200KB).
// ---------------------------------------------------------------------------
__global__ __launch_bounds__(BLK) void proposal_kernel(
    const float* __restrict__ anchors,   // (A, 4)
    const float* __restrict__ scores,    // (NB, A, 2), score = [...,0]
    const float* __restrict__ coeff,     // (NB, A, 4)
    float*       __restrict__ outScores, // (NB, POST_N)
    float*       __restrict__ outBoxes)  // (NB, POST_N, 4)
{
  const int b   = blockIdx.x;
  const int tid = threadIdx.x;
  const float* sb = scores + (size_t)b * A_COUNT * 2;

  __shared__ float    stage[DEPTH][BLK];        // 16 KB async staging ring
  __shared__ unsigned long long sbuf[SORT_N];   // 64 KB sort keys
  __shared__ uint32_t hist[256];
  __shared__ float    X1[PRE_N], Y1[PRE_N], X2[PRE_N], Y2[PRE_N], AREA[PRE_N];
  __shared__ uint32_t alive[NW];
  __shared__ int      keep[POST_N];
  __shared__ uint32_t sh_prefix, sh_need, sh_cnt;
  __shared__ int      sh_word;

  for (int i = tid; i < SORT_N; i += BLK) sbuf[i] = 0ull;
  if (tid == 0) { sh_prefix = 0u; sh_need = PRE_N; sh_cnt = 0u; sh_word = 0; }
  __syncthreads();

  // ---- 4-level radix select: T = PRE_N-th largest score-bit value ---------
  // (scores are uniform [0,1) => non-negative => float bits are monotonic)
  for (int lvl = 0; lvl < 4; ++lvl) {
    const int      shift = 24 - 8 * lvl;
    const uint32_t pmask = (lvl == 0) ? 0u : (0xFFFFFFFFu << (32 - 8 * lvl));
    for (int i = tid; i < 256; i += BLK) hist[i] = 0u;
    __syncthreads();
    const uint32_t prefix = sh_prefix;

    stream_scores(sb, stage, tid, [&](int g, uint32_t u) {
      (void)g;
      if ((u & pmask) == prefix)
        atomicAdd(&hist[(u >> shift) & 0xFF], 1u);
    });
    __syncthreads();

    if (tid == 0) {
      const uint32_t need = sh_need;
      uint32_t cum = 0u;
      for (int d = 255; d >= 0; --d) {
        const uint32_t c = hist[d];
        if (cum + c >= need) {
          sh_need   = need - cum;
          sh_prefix = prefix | ((uint32_t)d << shift);
          break;
        }
        cum += c;
      }
    }
    __syncthreads();
  }

  // ---- compact every element with bits >= T (count in [6000, 6000+ties)) --
  const uint32_t T = sh_prefix;
  stream_scores(sb, stage, tid, [&](int g, uint32_t u) {
    if (u >= T) {
      const uint32_t pos = atomicAdd(&sh_cnt, 1u);
      if (pos < SORT_N) {
        // key: score bits desc; among equal scores, lower index sorts first
        sbuf[pos] = ((unsigned long long)u << 32) |
                    (unsigned long long)(0xFFFFFFFFu - (uint32_t)g);
      }
    }
  });
  __syncthreads();

  // ---- LDS bitonic sort, descending, 8192 keys (zeros pad the tail) -------
  for (int k = 2; k <= SORT_N; k <<= 1) {
    for (int j = k >> 1; j > 0; j >>= 1) {
      for (int i = tid; i < SORT_N; i += BLK) {
        const int ixj = i ^ j;
        if (ixj > i) {
          const unsigned long long a = sbuf[i];
          const unsigned long long c = sbuf[ixj];
          const bool desc   = ((i & k) == 0);
          const bool doSwap = desc ? (a < c) : (a > c);
          if (doSwap) { sbuf[i] = c; sbuf[ixj] = a; }
        }
      }
      __syncthreads();
    }
  }

  // ---- decode + clip the sorted top-6000 boxes straight into LDS ----------
  const float4* anc4 = (const float4*)anchors;
  const float4* cb4  = (const float4*)(coeff + (size_t)b * A_COUNT * 4);
  for (int r = tid; r < PRE_N; r += BLK) {
    const uint32_t idx = 0xFFFFFFFFu - (uint32_t)sbuf[r];
    const float4 a = anc4[idx];
    const float4 d = cb4[idx];
    const float w  = a.z - a.x + 1.0f;
    const float h  = a.w - a.y + 1.0f;
    const float cx = a.x + 0.5f * w;
    const float cy = a.y + 0.5f * h;
    const float pcx = d.x * w + cx;
    const float pcy = d.y * h + cy;
    const float pw  = expf(d.z) * w;
    const float ph  = expf(d.w) * h;
    float x1 = pcx - 0.5f * pw;
    float y1 = pcy - 0.5f * ph;
    float x2 = pcx + 0.5f * pw;
    float y2 = pcy + 0.5f * ph;
    x1 = fminf(fmaxf(x1, 0.0f), IMG_MAX);
    y1 = fminf(fmaxf(y1, 0.0f), IMG_MAX);
    x2 = fminf(fmaxf(x2, 0.0f), IMG_MAX);
    y2 = fminf(fmaxf(y2, 0.0f), IMG_MAX);
    X1[r] = x1; Y1[r] = y1; X2[r] = x2; Y2[r] = y2;
    AREA[r] = (x2 - x1 + 1.0f) * (y2 - y1 + 1.0f);
  }
  for (int w = tid; w < NW; w += BLK)
    alive[w] = (w < NW - 1) ? 0xFFFFFFFFu : 0x0000FFFFu;  // 6000 = 187*32 + 16
  __syncthreads();

  // ---- 300 strictly sequential greedy NMS steps ---------------------------
  for (int step = 0; step < POST_N; ++step) {
    // All threads redundantly find the first alive index (same result; the
    // hint word is monotone non-decreasing since bits are only ever cleared).
    int w = sh_word;
    while (w < NW && alive[w] == 0u) ++w;
    const bool exh = (w == NW);                       // argmax(all False) == 0
    const int  i   = exh ? 0 : ((w << 5) + (__ffs(alive[w]) - 1));
    __syncthreads();   // everyone done reading alive/sh_word before writes

    if (tid == 0) { sh_word = w; keep[step] = i; }
    if (!exh) {        // uniform across the block
      const float bx1 = X1[i], by1 = Y1[i], bx2 = X2[i], by2 = Y2[i];
      const float ba  = AREA[i];
      for (int j = tid; j < PRE_N; j += BLK) {        // ~6 elements/thread
        const uint32_t m = 1u << (j & 31);
        if (alive[j >> 5] & m) {
          const float xx1 = fmaxf(bx1, X1[j]);
          const float yy1 = fmaxf(by1, Y1[j]);
          const float xx2 = fminf(bx2, X2[j]);
          const float yy2 = fminf(by2, Y2[j]);
          const float iw  = fmaxf(xx2 - xx1 + 1.0f, 0.0f);
          const float ih  = fmaxf(yy2 - yy1 + 1.0f, 0.0f);
          const float inter = iw * ih;
          const float iou   = inter / (ba + AREA[j] - inter);
          if (iou > NMS_THR)                       // clears i itself (IoU=1)
            atomicAnd(&alive[j >> 5], ~m);         // idempotent => determinstic
        }
      }
    }
    __syncthreads();   // suppression visible before next step's scan
  }

  // ---- gather outputs: scores (NB,POST_N,1) then boxes (NB,POST_N,4) ------
  for (int t = tid; t < POST_N; t += BLK) {
    const int i = keep[t];
    outScores[(size_t)b * POST_N + t] =
        __uint_as_float((uint32_t)(sbuf[i] >> 32));      // top_scores[i]
    const size_t ob = ((size_t)b * POST_N + t) * 4;
    outBoxes[ob + 0] = X1[i];
    outBoxes[ob + 1] = Y1[i];
    outBoxes[ob + 2] = X2[i];
    outBoxes[ob + 3] = Y2[i];
  }
}

// ---------------------------------------------------------------------------
extern "C" void kernel_launch(void* const* d_in, const int* in_sizes, int n_in,
                              void* d_out, int out_size, void* d_ws, size_t ws_size,
                              hipStream_t stream) {
  const float* anchors    = (const float*)d_in[0];  // (90000, 4)
  const float* bbox_score = (const float*)d_in[1];  // (32, 90000, 2)
  const float* bbox_coeff = (const float*)d_in[2];  // (32, 90000, 4)

  float* outScores = (float*)d_out;                       // 32*300
  float* outBoxes  = (float*)d_out + (size_t)NB * POST_N; // 32*300*4

  (void)d_ws; (void)ws_size; (void)in_sizes; (void)n_in; (void)out_size;

  proposal_kernel<<<NB, BLK, 0, stream>>>(anchors, bbox_score, bbox_coeff,
                                          outScores, outBoxes);
}